// CAVIEngine_15857019257119
// MI455X (gfx1250) — compile-verified
//
#include <hip/hip_runtime.h>
#include <stdint.h>

// ---------------- problem constants ----------------
#define BVAL 16            // batch (lanes per factor)
#define ITERS 5
#define DAMPING 0.5f
#define TPB 256            // 8 waves, 16 half-waves
#define CHUNK 1024         // factors per TDM chunk
#define SCATTER_BLOCKS 512

typedef unsigned int u32;
typedef unsigned long long u64;
typedef __attribute__((ext_vector_type(4))) u32 u32x4;
typedef __attribute__((ext_vector_type(8))) int i32x8;
typedef __attribute__((ext_vector_type(4))) int i32x4;

// sigmoid with fast hw reciprocal (v_exp_f32 + v_rcp_f32), ~1 ulp
__device__ __forceinline__ float fast_sigmoid(float x) {
    return __builtin_amdgcn_rcpf(1.0f + __expf(-x));
}

// Device-scope non-returning f32 atomic add, saddr form:
// 32-bit byte offset in VGPR + uniform 64-bit base in SGPR pair.
__device__ __forceinline__ void atomic_add_f32_dev(float* base, u32 byte_off, float v) {
    asm volatile("global_atomic_add_f32 %0, %1, %2 scope:SCOPE_DEV"
                 :: "v"(byte_off), "v"(v), "s"(base) : "memory");
}

// TDM: DMA `nwords` consecutive DWORDs from global memory into LDS at lds_off.
// D# group0: count=1, lds_addr, global_addr[56:0], type=2 ("image").
// D# group1: data_size=4B (code 2), tensor_dim0 = tile_dim0 = nwords (1-D tile),
//            tensor_dim1 = 1, tile_dim1/2 = 0 (unused), stride0 = nwords.
__device__ __forceinline__ void tdm_copy_words(u32 lds_off, const void* gptr, u32 nwords) {
    u64 ga = (u64)gptr;
    u32x4 g0;
    g0.x = 1u;                                              // count=1, user mode
    g0.y = lds_off;                                         // lds_addr (bytes)
    g0.z = (u32)ga;                                         // global_addr[31:0]
    g0.w = (u32)((ga >> 32) & 0x01FFFFFFu) | (2u << 30);    // addr[56:32] | type=2
    i32x8 g1;
    g1[0] = (int)(2u << 16);                                // data_size = 4 bytes
    g1[1] = (int)((nwords & 0xFFFFu) << 16);                // tensor_dim0[15:0]
    g1[2] = (int)(((nwords >> 16) & 0xFFFFu) | (1u << 16)); // tensor_dim0[31:16] | tensor_dim1=1
    g1[3] = (int)((nwords & 0xFFFFu) << 16);                // tile_dim0 = nwords
    g1[4] = 0;                                              // tile_dim1=0, tile_dim2=0
    g1[5] = (int)nwords;                                    // tensor_dim0_stride lo
    g1[6] = 0;
    g1[7] = 0;
    i32x4 z4 = {0, 0, 0, 0};
    i32x8 z8 = {0, 0, 0, 0, 0, 0, 0, 0};
    __builtin_amdgcn_tensor_load_to_lds(g0, g1, z4, z4, z8, 0);
}

// ---------------- init: transpose evidence, sigmoid, zero delta (float4/thread) ----------------
__global__ void __launch_bounds__(TPB)
init_kernel(const float* __restrict__ ev, float* __restrict__ logits_t,
            float* __restrict__ qt, float* __restrict__ delta, int N) {
    int t = (int)blockIdx.x * TPB + (int)threadIdx.x;    // handles elements [4t, 4t+4)
    if (t >= N * (BVAL / 4)) return;
    int n = t >> 2;
    int bb = (t & 3) * 4;
    float4 e;
    e.x = ev[(size_t)(bb + 0) * N + n];
    e.y = ev[(size_t)(bb + 1) * N + n];
    e.z = ev[(size_t)(bb + 2) * N + n];
    e.w = ev[(size_t)(bb + 3) * N + n];
    ((float4*)logits_t)[t] = e;
    float4 q;
    q.x = fast_sigmoid(e.x);
    q.y = fast_sigmoid(e.y);
    q.z = fast_sigmoid(e.z);
    q.w = fast_sigmoid(e.w);
    ((float4*)qt)[t] = q;
    ((float4*)delta)[t] = make_float4(0.f, 0.f, 0.f, 0.f);
}

// ---------------- scatter: half-wave per factor, TDM-staged factor stream ----------------
// MODE 1: t1 (2 indices), MODE 2: t2, MODE 3: t3 (3 indices each)
template <int MODE>
__global__ void __launch_bounds__(TPB)
scatter_kernel(const int* __restrict__ idx, const float* __restrict__ wts,
               const float* __restrict__ qt, float* __restrict__ delta, int R) {
    constexpr int IPF = (MODE == 1) ? 2 : 3;
    __shared__ int   s_idx[2][CHUNK * IPF];
    __shared__ float s_w[2][CHUNK];

    const int span = (R + (int)gridDim.x - 1) / (int)gridDim.x;
    const int f0 = (int)blockIdx.x * span;
    const int f1 = (f0 + span < R) ? (f0 + span) : R;
    if (f0 >= f1) return;
    const int nchunks = (f1 - f0 + CHUNK - 1) / CHUNK;

    // prologue: stage chunk 0 into buffer 0 (TDM issue by wave 0)
    if (threadIdx.x == 0) {
        int cc = (f1 - f0 < CHUNK) ? (f1 - f0) : CHUNK;
        tdm_copy_words((u32)(uintptr_t)&s_idx[0][0], idx + (size_t)f0 * IPF, (u32)(cc * IPF));
        tdm_copy_words((u32)(uintptr_t)&s_w[0][0],   wts + f0,               (u32)cc);
    }

    const int lane = (int)threadIdx.x & (BVAL - 1);  // batch element
    const int hw   = (int)threadIdx.x / BVAL;        // half-wave id 0..15
    const int nhw  = TPB / BVAL;
    float* dl = delta;

    for (int ci = 0; ci < nchunks; ++ci) {
        if (threadIdx.x == 0) {
            if (ci + 1 < nchunks) {  // issue next chunk, then wait for current (2 TDM ops in flight)
                int nf0 = f0 + (ci + 1) * CHUNK;
                int cc = (f1 - nf0 < CHUNK) ? (f1 - nf0) : CHUNK;
                int buf = (ci + 1) & 1;
                tdm_copy_words((u32)(uintptr_t)&s_idx[buf][0], idx + (size_t)nf0 * IPF, (u32)(cc * IPF));
                tdm_copy_words((u32)(uintptr_t)&s_w[buf][0],   wts + nf0,               (u32)cc);
                __builtin_amdgcn_s_wait_tensorcnt(2);
            } else {
                __builtin_amdgcn_s_wait_tensorcnt(0);
            }
        }
        __syncthreads();  // TDM data for chunk ci visible to all waves

        const int buf = ci & 1;
        const int cf0 = f0 + ci * CHUNK;
        const int cc = (f1 - cf0 < CHUNK) ? (f1 - cf0) : CHUNK;
        const int* si = &s_idx[buf][0];
        const float* sw = &s_w[buf][0];

        for (int fl = hw; fl < cc; fl += nhw) {
            float w = sw[fl];  // LDS broadcast across the half-wave
            if (MODE == 1) {
                int a = si[2 * fl + 0];
                int c = si[2 * fl + 1];
                int oa = a * BVAL + lane;                 // 32-bit offsets; arrays < 16MB
                int oc = c * BVAL + lane;
                float qa = qt[oa];                        // one 64B line per gather
                float qc = qt[oc];
                atomic_add_f32_dev(dl, (u32)oc * 4u, w * qa);
                atomic_add_f32_dev(dl, (u32)oa * 4u, w * (qc - 1.0f));
            } else {
                int a = si[3 * fl + 0];
                int b = si[3 * fl + 1];
                int c = si[3 * fl + 2];
                int oa = a * BVAL + lane;
                int ob_ = b * BVAL + lane;
                int oc = c * BVAL + lane;
                float qa = qt[oa];
                float qb = qt[ob_];
                float qc = qt[oc];
                float dc, da, db;
                if (MODE == 2) {
                    dc = w * qa * qb;
                    da = w * qb * (qc - 1.0f);
                    db = w * qa * (qc - 1.0f);
                } else {
                    float onb = 1.0f - qb;
                    dc = w * qa * onb;
                    da = w * onb * (qc - 1.0f);
                    db = w * qa * (1.0f - qc);
                }
                atomic_add_f32_dev(dl, (u32)oc * 4u, dc);
                atomic_add_f32_dev(dl, (u32)oa * 4u, da);
                atomic_add_f32_dev(dl, (u32)ob_ * 4u, db);
            }
        }
        __syncthreads();  // done reading buffer before it is refilled
    }
}

// ---------------- update: damped logit update + sigmoid; re-zero delta (float4/thread) ----------------
__global__ void __launch_bounds__(TPB)
update_kernel(const float* __restrict__ ev, float* __restrict__ logits_t,
              float* __restrict__ qt, float* __restrict__ delta,
              float* __restrict__ out, int N, int write_out) {
    int t = (int)blockIdx.x * TPB + (int)threadIdx.x;    // handles elements [4t, 4t+4)
    if (t >= N * (BVAL / 4)) return;
    int n = t >> 2;
    int bb = (t & 3) * 4;
    float4 e;
    e.x = ev[(size_t)(bb + 0) * N + n];
    e.y = ev[(size_t)(bb + 1) * N + n];
    e.z = ev[(size_t)(bb + 2) * N + n];
    e.w = ev[(size_t)(bb + 3) * N + n];
    float4 l = ((float4*)logits_t)[t];
    float4 d = ((float4*)delta)[t];
    float4 nl;  // target = e + total; damped mix
    nl.x = (1.0f - DAMPING) * l.x + DAMPING * (e.x + d.x);
    nl.y = (1.0f - DAMPING) * l.y + DAMPING * (e.y + d.y);
    nl.z = (1.0f - DAMPING) * l.z + DAMPING * (e.z + d.z);
    nl.w = (1.0f - DAMPING) * l.w + DAMPING * (e.w + d.w);
    ((float4*)logits_t)[t] = nl;
    float4 q;
    q.x = fast_sigmoid(nl.x);
    q.y = fast_sigmoid(nl.y);
    q.z = fast_sigmoid(nl.z);
    q.w = fast_sigmoid(nl.w);
    ((float4*)qt)[t] = q;
    ((float4*)delta)[t] = make_float4(0.f, 0.f, 0.f, 0.f);  // ready for next iteration
    if (write_out) {
        out[(size_t)(bb + 0) * N + n] = q.x;
        out[(size_t)(bb + 1) * N + n] = q.y;
        out[(size_t)(bb + 2) * N + n] = q.z;
        out[(size_t)(bb + 3) * N + n] = q.w;
    }
}

extern "C" void kernel_launch(void* const* d_in, const int* in_sizes, int n_in,
                              void* d_out, int out_size, void* d_ws, size_t ws_size,
                              hipStream_t stream) {
    const float* ev = (const float*)d_in[0];
    const float* w1 = (const float*)d_in[1];
    const float* w2 = (const float*)d_in[2];
    const float* w3 = (const float*)d_in[3];
    const int* i1 = (const int*)d_in[4];
    const int* i2 = (const int*)d_in[5];
    const int* i3 = (const int*)d_in[6];

    const int N  = in_sizes[0] / BVAL;
    const int R1 = in_sizes[1];
    const int R2 = in_sizes[2];
    const int R3 = in_sizes[3];
    const int NB = N * BVAL;

    float* logits_t = (float*)d_ws;       // (N,16) transposed state, 64B per node
    float* qt       = logits_t + NB;
    float* delta    = qt + NB;

    const int egrid = (NB / 4 + TPB - 1) / TPB;
    init_kernel<<<egrid, TPB, 0, stream>>>(ev, logits_t, qt, delta, N);
    for (int it = 0; it < ITERS; ++it) {
        scatter_kernel<1><<<SCATTER_BLOCKS, TPB, 0, stream>>>(i1, w1, qt, delta, R1);
        scatter_kernel<2><<<SCATTER_BLOCKS, TPB, 0, stream>>>(i2, w2, qt, delta, R2);
        scatter_kernel<3><<<SCATTER_BLOCKS, TPB, 0, stream>>>(i3, w3, qt, delta, R3);
        update_kernel<<<egrid, TPB, 0, stream>>>(ev, logits_t, qt, delta,
                                                 (float*)d_out, N,
                                                 (it == ITERS - 1) ? 1 : 0);
    }
}